// mult_h_attention_82188494176271
// MI455X (gfx1250) — compile-verified
//
#include <hip/hip_runtime.h>

#define EMBED 1024
#define HEADS 16
#define HDIM  64
#define BATCH 4
#define SEQ   1024

typedef __attribute__((ext_vector_type(2))) float v2f;
typedef __attribute__((ext_vector_type(8))) float v8f;

// D = A(16x4 f32) * B(4x16 f32) + C(16x16 f32)   -- CDNA5 V_WMMA_F32_16X16X4_F32
static __device__ __forceinline__ v8f wmma_f32(v2f a, v2f b, v8f c) {
  return __builtin_amdgcn_wmma_f32_16x16x4_f32(
      /*neg_a=*/false, a, /*neg_b=*/false, b,
      /*c_mod=*/(short)0, c, /*reuse_a=*/false, /*reuse_b=*/false);
}

static __device__ __forceinline__ v8f vzero8() {
  return (v8f){0.f, 0.f, 0.f, 0.f, 0.f, 0.f, 0.f, 0.f};
}

// ---------------------------------------------------------------------------
// Kernel A: per-head linear projections.  X viewed as [B*S*H, 64] rows.
//   y = x @ W.T      (W is [64,64], shared across heads)
// tensor 0: query->Wq -> Qp [B,H,S,D]
// tensor 1: keys ->Wk -> Kp [B,H,S,D]
// tensor 2: values->Wv -> Vt [B,H,D,S]   (transposed for P*V B-fragments)
// One wave handles 16 rows (= the 16 heads of one token) x 64 output dims.
// ---------------------------------------------------------------------------
__global__ __launch_bounds__(128) void qkv_proj_kernel(
    const float* __restrict__ q_in, const float* __restrict__ k_in,
    const float* __restrict__ v_in,
    const float* __restrict__ Wq, const float* __restrict__ Wk,
    const float* __restrict__ Wv,
    float* __restrict__ Qp, float* __restrict__ Kp, float* __restrict__ Vt)
{
  const int wave   = blockIdx.x * 4 + (threadIdx.x >> 5);
  const int lane   = threadIdx.x & 31;
  const int m      = lane & 15;      // A-matrix row within tile
  const int half   = lane >> 4;      // 0: K-pair {0,1}, 1: K-pair {2,3}
  const int tensor = wave >> 12;     // 4096 row-tiles per tensor
  const int token  = wave & 4095;    // = b*SEQ + s
  const int b      = token >> 10;
  const int s      = token & 1023;

  const float* X;
  const float* W;
  if (tensor == 0)      { X = q_in; W = Wq; }
  else if (tensor == 1) { X = k_in; W = Wk; }
  else                  { X = v_in; W = Wv; }

  // A fragments: 16 rows (heads of this token) x 64 features
  const float* arow = X + (size_t)(token * 16 + m) * HDIM + 2 * half;
  v2f qa[16];
#pragma unroll
  for (int ks = 0; ks < 16; ++ks)
    qa[ks] = *(const v2f*)(arow + 4 * ks);

#pragma unroll
  for (int nt = 0; nt < 4; ++nt) {
    // B fragments, batched: B[k][n] = W[n][k]  (x @ W.T)
    const float* brow = W + (size_t)(nt * 16 + m) * HDIM + 2 * half;
    v2f bf[16];
#pragma unroll
    for (int ks = 0; ks < 16; ++ks)
      bf[ks] = *(const v2f*)(brow + 4 * ks);

    // two independent accumulator chains
    v8f a0 = vzero8(), a1 = vzero8();
#pragma unroll
    for (int ks = 0; ks < 8; ++ks) {
      a0 = wmma_f32(qa[2 * ks + 0], bf[2 * ks + 0], a0);
      a1 = wmma_f32(qa[2 * ks + 1], bf[2 * ks + 1], a1);
    }
    v8f acc = a0 + a1;

    // D layout: lane holds row (j + 8*half) = head, col (nt*16 + m) = dim.
    // Both output layouts have j-stride 65536 floats (h-stride).
    float* outp;
    size_t base;
    if (tensor == 2) {
      outp = Vt;
      base = ((size_t)((b * 16 + 8 * half) * HDIM + nt * 16 + m)) * SEQ + s;
    } else {
      outp = (tensor == 0) ? Qp : Kp;
      base = ((size_t)((b * 16 + 8 * half) * SEQ + s)) * HDIM + nt * 16 + m;
    }
#pragma unroll
    for (int j = 0; j < 8; ++j)
      outp[base + (size_t)j * (HDIM * SEQ)] = acc[j];
  }
}

// ---------------------------------------------------------------------------
// Kernel B: flash attention per (b,h).  One wave = one 16-query tile,
// streaming 64 key tiles of 16, online softmax, P*V accumulated in VGPRs.
// The 4 waves of a block share (b,h) and are barrier-locked per key tile,
// so K/V tiles are served from WGP$ after the first wave touches them.
// ---------------------------------------------------------------------------
__global__ __launch_bounds__(128) void attn_kernel(
    const float* __restrict__ Qp, const float* __restrict__ Kp,
    const float* __restrict__ Vt, const int* __restrict__ mask,
    float* __restrict__ AO)
{
  __shared__ float lds_p[4][16 * 18];   // padded stride 18: 8B aligned, no bank conflicts

  const int wslot = threadIdx.x >> 5;
  const int wave  = blockIdx.x * 4 + wslot;
  const int lane  = threadIdx.x & 31;
  const int m     = lane & 15;
  const int half  = lane >> 4;
  const int qt    = wave & 63;          // 64 query tiles per (b,h)
  const int bh    = wave >> 6;          // 0..63
  const int b     = bh >> 4;
  const int h     = bh & 15;
  const int qbase = qt * 16;

  const float* Qb = Qp + (size_t)bh * SEQ * HDIM;
  const float* Kb = Kp + (size_t)bh * SEQ * HDIM;
  const float* Vb = Vt + (size_t)bh * HDIM * SEQ;
  const int*   Mb = mask + (size_t)b * SEQ * SEQ;

  // Preload Q fragments for this 16-query tile (reused over all key tiles)
  const float* qrow = Qb + (size_t)(qbase + m) * HDIM + 2 * half;
  v2f qa[16];
#pragma unroll
  for (int ks = 0; ks < 16; ++ks)
    qa[ks] = *(const v2f*)(qrow + 4 * ks);

  v8f O[4];
  float rowmax[8], rowsum[8];
#pragma unroll
  for (int nt = 0; nt < 4; ++nt) O[nt] = vzero8();
#pragma unroll
  for (int j = 0; j < 8; ++j) { rowmax[j] = -3.4e38f; rowsum[j] = 0.f; }

  float* P = lds_p[wslot];
  const int* mrow0 = Mb + (size_t)(qbase + 8 * half) * SEQ + m;

  for (int kt = 0; kt < 64; ++kt) {
    const int kb0 = kt * 16;
    const float* krow = Kb + (size_t)(kb0 + m) * HDIM + 2 * half;

    // batch-load K fragments (distinct regs -> loads overlap the MMA chain)
    v2f kf[16];
#pragma unroll
    for (int ks = 0; ks < 16; ++ks)
      kf[ks] = *(const v2f*)(krow + 4 * ks);

    // prefetch next key tile of K/V into caches while we compute
    if (kt < 63) {
      __builtin_prefetch(krow + 16 * HDIM, 0, 1);
      __builtin_prefetch(Vb + (size_t)m * SEQ + kb0 + 16, 0, 1);
    }

    // batch-load mask for this tile (j-stride = SEQ -> immediate offsets)
    int mk[8];
#pragma unroll
    for (int j = 0; j < 8; ++j)
      mk[j] = mrow0[(size_t)j * SEQ + kb0];

    // ---- scores: S = Q (16x64) @ K_tile^T (64x16), two chains ----
    v8f c0 = vzero8(), c1 = vzero8();
#pragma unroll
    for (int ks = 0; ks < 8; ++ks) {
      c0 = wmma_f32(qa[2 * ks + 0], kf[2 * ks + 0], c0);
      c1 = wmma_f32(qa[2 * ks + 1], kf[2 * ks + 1], c1);
    }
    v8f c = c0 + c1;

    // ---- mask + scale + online softmax (row = query, along 16 lanes) ----
    float p[8], scl[8];
#pragma unroll
    for (int j = 0; j < 8; ++j) {
      // reference: where(mask==0, -1e20) happens BEFORE /sqrt(EMBED)
      const float sv = (mk[j] == 0) ? -3.125e18f : c[j] * 0.03125f;

      float mt = sv;                           // max over the 16 keys
      mt = fmaxf(mt, __shfl_xor(mt, 1, 16));
      mt = fmaxf(mt, __shfl_xor(mt, 2, 16));
      mt = fmaxf(mt, __shfl_xor(mt, 4, 16));
      mt = fmaxf(mt, __shfl_xor(mt, 8, 16));
      const float nm = fmaxf(rowmax[j], mt);
      const float sc = __expf(rowmax[j] - nm);
      rowmax[j] = nm;
      const float pv = __expf(sv - nm);
      float st = pv;                           // sum over the 16 keys
      st += __shfl_xor(st, 1, 16);
      st += __shfl_xor(st, 2, 16);
      st += __shfl_xor(st, 4, 16);
      st += __shfl_xor(st, 8, 16);
      rowsum[j] = rowsum[j] * sc + st;
      p[j]   = pv;
      scl[j] = sc;
    }
#pragma unroll
    for (int nt = 0; nt < 4; ++nt)
#pragma unroll
      for (int j = 0; j < 8; ++j)
        O[nt][j] *= scl[j];

    // ---- stage P (C-layout -> A-layout) through LDS ----
    __syncthreads();                 // previous iteration's reads are done
#pragma unroll
    for (int j = 0; j < 8; ++j)
      P[(j + 8 * half) * 18 + m] = p[j];       // P[query][key]
    __syncthreads();
    v2f pa[4];
#pragma unroll
    for (int ks = 0; ks < 4; ++ks)
      pa[ks] = *(const v2f*)(P + m * 18 + 4 * ks + 2 * half);

    // ---- O += P (16x16) @ V_tile (16x64) ----
    // batch-load all 16 V fragments first, then 16 WMMAs on 4 indep chains
    v2f vf[4][4];
#pragma unroll
    for (int nt = 0; nt < 4; ++nt) {
      const float* vrow = Vb + (size_t)(nt * 16 + m) * SEQ + kb0 + 2 * half;
#pragma unroll
      for (int ks = 0; ks < 4; ++ks)
        vf[nt][ks] = *(const v2f*)(vrow + 4 * ks);
    }
#pragma unroll
    for (int ks = 0; ks < 4; ++ks)
#pragma unroll
      for (int nt = 0; nt < 4; ++nt)
        O[nt] = wmma_f32(pa[ks], vf[nt][ks], O[nt]);
  }

  // finalize: divide by rowsum, store as [B, S, H*D]  (j-stride = 1024 floats)
  float inv[8];
#pragma unroll
  for (int j = 0; j < 8; ++j) inv[j] = 1.f / rowsum[j];
#pragma unroll
  for (int nt = 0; nt < 4; ++nt) {
    const size_t base =
        ((size_t)(b * SEQ + qbase + 8 * half) * HEADS + h) * HDIM + nt * 16 + m;
#pragma unroll
    for (int j = 0; j < 8; ++j)
      AO[base + (size_t)j * EMBED] = O[nt][j] * inv[j];
  }
}

// ---------------------------------------------------------------------------
// Kernel C: out = X[4096,1024] @ Wu^T + bu.  Wave = 32x64 output tile
// (two 16-row M-tiles share the same Wu B-fragments: halves Wu traffic,
//  32 WMMAs per 24 loads per K-block, 8 independent accumulator chains).
// ---------------------------------------------------------------------------
__global__ __launch_bounds__(128) void out_proj_kernel(
    const float* __restrict__ X, const float* __restrict__ Wu,
    const float* __restrict__ bu, float* __restrict__ out)
{
  const int wave  = blockIdx.x * 4 + (threadIdx.x >> 5);
  const int lane  = threadIdx.x & 31;
  const int m     = lane & 15;
  const int half  = lane >> 4;
  const int ntile = wave & 15;          // 16 tiles of 64 output cols
  const int mtile = wave >> 4;          // 128 tiles of 32 rows
  const int row0  = mtile * 32;
  const int col0  = ntile * 64;

  v8f acc[2][4];
#pragma unroll
  for (int mt = 0; mt < 2; ++mt)
#pragma unroll
    for (int nt = 0; nt < 4; ++nt) acc[mt][nt] = vzero8();

  const float* arow0 = X  + (size_t)(row0 + m) * EMBED + 2 * half;
  const float* arow1 = arow0 + (size_t)16 * EMBED;
  const float* brow  = Wu + (size_t)(col0 + m) * EMBED + 2 * half;

  for (int kb = 0; kb < 64; ++kb) {   // 64 blocks x 4 k-steps = K=1024
    v2f af[2][4];
    v2f bf[4][4];
#pragma unroll
    for (int i = 0; i < 4; ++i) {
      af[0][i] = *(const v2f*)(arow0 + 16 * kb + 4 * i);
      af[1][i] = *(const v2f*)(arow1 + 16 * kb + 4 * i);
    }
#pragma unroll
    for (int nt = 0; nt < 4; ++nt)
#pragma unroll
      for (int i = 0; i < 4; ++i)
        bf[nt][i] = *(const v2f*)(brow + (size_t)nt * 16 * EMBED + 16 * kb + 4 * i);

#pragma unroll
    for (int i = 0; i < 4; ++i)
#pragma unroll
      for (int mt = 0; mt < 2; ++mt)
#pragma unroll
        for (int nt = 0; nt < 4; ++nt)
          acc[mt][nt] = wmma_f32(af[mt][i], bf[nt][i], acc[mt][nt]);
  }

#pragma unroll
  for (int mt = 0; mt < 2; ++mt)
#pragma unroll
    for (int nt = 0; nt < 4; ++nt) {
      const float bias = bu[col0 + nt * 16 + m];
      const size_t base =
          (size_t)(row0 + mt * 16 + 8 * half) * EMBED + col0 + nt * 16 + m;
#pragma unroll
      for (int j = 0; j < 8; ++j)
        out[base + (size_t)j * EMBED] = acc[mt][nt][j] + bias;
    }
}

// ---------------------------------------------------------------------------
extern "C" void kernel_launch(void* const* d_in, const int* in_sizes, int n_in,
                              void* d_out, int out_size, void* d_ws, size_t ws_size,
                              hipStream_t stream) {
  (void)in_sizes; (void)n_in; (void)out_size; (void)ws_size;
  const float* values = (const float*)d_in[0];
  const float* keys   = (const float*)d_in[1];
  const float* query  = (const float*)d_in[2];
  const int*   mask   = (const int*)d_in[3];
  const float* Wv     = (const float*)d_in[4];
  const float* Wq     = (const float*)d_in[5];
  const float* Wk     = (const float*)d_in[6];
  const float* Wu     = (const float*)d_in[7];
  const float* bu     = (const float*)d_in[8];
  float* out = (float*)d_out;

  // workspace layout (f32): Qp | Kp | Vt | AO, 4M floats each = 64 MB total
  float* ws = (float*)d_ws;
  const size_t T = (size_t)BATCH * HEADS * SEQ * HDIM;   // 4,194,304
  float* Qp = ws;
  float* Kp = ws + T;
  float* Vt = ws + 2 * T;
  float* AO = ws + 3 * T;

  // A: 3 tensors * 4096 row-tiles = 12288 waves / 4 per block
  qkv_proj_kernel<<<3072, 128, 0, stream>>>(query, keys, values, Wq, Wk, Wv,
                                            Qp, Kp, Vt);
  // B: 64 (b,h) * 64 q-tiles = 4096 waves / 4 per block
  attn_kernel<<<1024, 128, 0, stream>>>(Qp, Kp, Vt, mask, AO);
  // C: 128 m-tiles(32 rows) * 16 n-tiles = 2048 waves / 4 per block
  out_proj_kernel<<<512, 128, 0, stream>>>(AO, Wu, bu, out);
}